// CTC_54992761258536
// MI455X (gfx1250) — compile-verified
//
#include <hip/hip_runtime.h>
#include <hip/hip_bf16.h>

// CTC loss for input [T=256, B=32, C=6625] f32, label [32, 25] i32.
// Memory-bound: one 217.6 MB streaming pass (HBM floor ~9.4us @ 23.3 TB/s).
// WMMA f32 16x16x4 with an all-ones B performs the sum-exp row reduction in
// the matrix pipe while VALU does the exps. Hot loop is branchless so loads
// pipeline (no per-load s_wait_loadcnt 0); 4x unroll with 4 independent
// accumulators breaks the WMMA D->C dependency chain.

#define Tn 256
#define Bn 32
#define Cn 6625
#define Ln 25
#define Sn 51            // 2*Ln + 1
#define NSEG 16
#define SEGLEN 415       // ceil(6625/16)
#define NEGF (-1e9f)

typedef float __attribute__((ext_vector_type(2))) v2f;
typedef float __attribute__((ext_vector_type(8))) v8f;

__device__ __forceinline__ v8f wmma_rowsum(v2f a, v2f ones, v8f c) {
    return __builtin_amdgcn_wmma_f32_16x16x4_f32(
        /*neg_a=*/false, a, /*neg_b=*/false, ones,
        /*c_mod=*/(short)0, c, /*reuse_a=*/false, /*reuse_b=*/false);
}

// ---------------------------------------------------------------------------
// Kernel 0: zero the row-sum accumulators (ws) and the scalar output.
// ---------------------------------------------------------------------------
__global__ __launch_bounds__(256) void ctc_init(float* __restrict__ rowsum,
                                                float* __restrict__ out) {
    int i = blockIdx.x * blockDim.x + threadIdx.x;
    if (i < Tn * Bn) rowsum[i] = 0.0f;
    if (i == 0) out[0] = 0.0f;
}

// ---------------------------------------------------------------------------
// Kernel 1: per-row sum of exp(x). One wave handles 16 rows x 1 of 16 column
// segments. A-matrix (16x4 f32) per ISA layout: lanes 0-15 carry M=lane with
// K={0,1}; lanes 16-31 carry M=lane-16 with K={2,3}. B = ones so
// D[m,n] += sum_k A[m,k]: the WMMA accumulators hold running row sums.
// ---------------------------------------------------------------------------
__global__ __launch_bounds__(256) void ctc_sumexp(const float* __restrict__ in,
                                                  float* __restrict__ rowsum) {
    const int lane   = threadIdx.x & 31;
    const int waveIb = threadIdx.x >> 5;
    const int gw     = blockIdx.x * 8 + waveIb;   // 0..8191 global wave id
    const int group  = gw >> 4;                   // 0..511 : 16-row group
    const int seg    = gw & 15;                   // 0..15  : column segment

    const int m    = lane & 15;
    const int kofs = (lane >> 4) << 1;            // 0 for lanes<16, 2 otherwise
    const int row  = group * 16 + m;              // 0..8191 == t*Bn + b
    const float* rp = in + (size_t)row * Cn;

    const int c0   = seg * SEGLEN;
    const int cend = min(c0 + SEGLEN, Cn);
    const int nfull = (cend - c0) >> 2;           // unguarded 4-col chunks

    v2f ones; ones.x = 1.0f; ones.y = 1.0f;
    v8f acc0 = {}, acc1 = {}, acc2 = {}, acc3 = {};

    const float* p = rp + c0 + kofs;              // this lane's first column
    int i = 0;
    // Steady state: branchless, 8 independent loads issued before use,
    // 4 WMMAs into independent accumulators.
    for (; i + 4 <= nfull; i += 4, p += 16) {
        float x0 = p[0],  x1 = p[1];
        float x2 = p[4],  x3 = p[5];
        float x4 = p[8],  x5 = p[9];
        float x6 = p[12], x7 = p[13];
        v2f a0; a0.x = __expf(x0); a0.y = __expf(x1);
        v2f a1; a1.x = __expf(x2); a1.y = __expf(x3);
        v2f a2; a2.x = __expf(x4); a2.y = __expf(x5);
        v2f a3; a3.x = __expf(x6); a3.y = __expf(x7);
        acc0 = wmma_rowsum(a0, ones, acc0);
        acc1 = wmma_rowsum(a1, ones, acc1);
        acc2 = wmma_rowsum(a2, ones, acc2);
        acc3 = wmma_rowsum(a3, ones, acc3);
    }
    // Remaining full chunks (0..3 of them).
    for (; i < nfull; ++i, p += 4) {
        float x0 = p[0], x1 = p[1];
        v2f a; a.x = __expf(x0); a.y = __expf(x1);
        acc0 = wmma_rowsum(a, ones, acc0);
    }
    // Guarded tail chunk (uniform branch per wave; 0..3 columns).
    int ctail = c0 + (nfull << 2);
    if (ctail < cend) {
        int col = ctail + kofs;
        float x0 = (col     < cend) ? rp[col]     : -1e30f;  // exp -> 0
        float x1 = (col + 1 < cend) ? rp[col + 1] : -1e30f;
        v2f a; a.x = __expf(x0); a.y = __expf(x1);
        acc0 = wmma_rowsum(a, ones, acc0);
    }

    v8f acc = (acc0 + acc1) + (acc2 + acc3);

    // C/D layout: lane n<16 holds rows 0..7 in VGPR 0..7; lane>=16 rows 8..15.
    if (lane == 0) {
#pragma unroll
        for (int j = 0; j < 8; ++j) atomicAdd(&rowsum[group * 16 + j], acc[j]);
    }
    if (lane == 16) {
#pragma unroll
        for (int j = 0; j < 8; ++j) atomicAdd(&rowsum[group * 16 + 8 + j], acc[j]);
    }
}

// ---------------------------------------------------------------------------
// Kernel 2: lse[r] = log(sum_exp[r]) in place.
// ---------------------------------------------------------------------------
__global__ __launch_bounds__(256) void ctc_lse(float* __restrict__ rowsum) {
    int i = blockIdx.x * blockDim.x + threadIdx.x;
    if (i < Tn * Bn) rowsum[i] = __logf(rowsum[i]);
}

// ---------------------------------------------------------------------------
// Kernel 3: forward DP. One 64-thread block per batch; lp_ext staged in LDS,
// 255-step double-buffered logaddexp recurrence over S=51 states.
// ---------------------------------------------------------------------------
__global__ __launch_bounds__(64) void ctc_dp(const float* __restrict__ in,
                                             const int* __restrict__ label,
                                             const float* __restrict__ lse,
                                             float* __restrict__ out) {
    __shared__ int   s_ext[Sn];
    __shared__ int   s_skip[Sn];
    __shared__ int   s_len;
    __shared__ float s_lp[Tn * Sn];     // 52224 B
    __shared__ float s_a[Sn];
    __shared__ float s_b[Sn];

    const int b   = blockIdx.x;
    const int tid = threadIdx.x;

    if (tid == 0) {
        int packed[Ln];
        int n = 0;
        for (int i = 0; i < Ln; ++i) {            // stable front-pack of nonzeros
            int v = label[b * Ln + i];
            if (v != 0) packed[n++] = v;
        }
        s_len = n;
        for (int i = n; i < Ln; ++i) packed[i] = 0;
        for (int s = 0; s < Sn; ++s) s_ext[s] = 0;
        for (int i = 0; i < Ln; ++i) s_ext[2 * i + 1] = packed[i];
        for (int s = 0; s < Sn; ++s)
            s_skip[s] = (s >= 2) && (s_ext[s] != 0) && (s_ext[s] != s_ext[s - 2]);
    }
    __syncthreads();

    // Stage lp_ext[t][s] = x[t,b,ext[s]] - lse[t,b] (independent, pipelined loads)
    for (int i = tid; i < Tn * Sn; i += 64) {
        int t = i / Sn;
        int s = i - t * Sn;
        s_lp[i] = in[((size_t)t * Bn + b) * Cn + s_ext[s]] - lse[t * Bn + b];
    }
    __syncthreads();

    if (tid < Sn) s_a[tid] = (tid < 2) ? s_lp[tid] : NEGF;
    __syncthreads();

    float* cur = s_a;
    float* nxt = s_b;
    for (int t = 1; t < Tn; ++t) {
        if (tid < Sn) {
            float a1 = cur[tid];
            float a2 = (tid >= 1) ? cur[tid - 1] : NEGF;
            float a3 = (tid >= 2 && s_skip[tid]) ? cur[tid - 2] : NEGF;
            float mx = fmaxf(a1, fmaxf(a2, a3));
            float sm = __expf(a1 - mx) + __expf(a2 - mx) + __expf(a3 - mx);
            nxt[tid] = mx + __logf(sm) + s_lp[t * Sn + tid];
        }
        __syncthreads();
        float* tmp = cur; cur = nxt; nxt = tmp;
    }

    if (tid == 0) {
        int len = s_len;
        int i1 = 2 * len - 1; if (i1 < 0) i1 = 0;
        int i2 = 2 * len;     if (i2 > Sn - 1) i2 = Sn - 1;
        float e1 = cur[i1], e2 = cur[i2];
        float mx = fmaxf(e1, e2);
        float lsum = mx + __logf(__expf(e1 - mx) + __expf(e2 - mx));
        float loss = -lsum;
        if (loss > 1e8f) loss = 0.0f;               // zero_infinity
        atomicAdd(out, loss / (float)(Bn * Tn));    // mean over B, then / T
    }
}

// ---------------------------------------------------------------------------
extern "C" void kernel_launch(void* const* d_in, const int* in_sizes, int n_in,
                              void* d_out, int out_size, void* d_ws, size_t ws_size,
                              hipStream_t stream) {
    const float* input = (const float*)d_in[0];   // [256, 32, 6625] f32
    const int*   label = (const int*)d_in[1];     // [32, 25] i32
    float*       out   = (float*)d_out;           // scalar f32
    float*       rowsum = (float*)d_ws;           // 8192 floats (32 KB scratch)

    ctc_init<<<(Tn * Bn + 255) / 256, 256, 0, stream>>>(rowsum, out);
    ctc_sumexp<<<(Tn * Bn / 16) * NSEG / 8, 256, 0, stream>>>(input, rowsum);
    ctc_lse<<<(Tn * Bn + 255) / 256, 256, 0, stream>>>(rowsum);
    ctc_dp<<<Bn, 64, 0, stream>>>(input, label, rowsum, out);
}